// GraphSAGE_70068096467620
// MI455X (gfx1250) — compile-verified
//
#include <hip/hip_runtime.h>

// Problem constants (from reference): B=8, C=128, N=10000, K=16, COUT=256
#define BB 8
#define CC 128
#define NN 10000
#define KK 16
#define COUT 256

typedef __attribute__((ext_vector_type(2))) float v2f;
typedef __attribute__((ext_vector_type(8))) float v8f;

// ---------------------------------------------------------------------------
// WMMA f32 16x16x4 lane mapping (CDNA5 ISA 7.12.2, wave32):
//   A (16x4, MxK): lane L holds row m = L&15; VGPR v holds k = 2*(L>>4) + v
//   B (4x16, KxN): lane L holds col n = L&15; VGPR v holds k = 2*(L>>4) + v
//   C/D (16x16):   lane L holds col n = L&15; VGPR v holds row m = v + 8*(L>>4)
// EXEC must be all-ones around the WMMA: all code below is convergent.
// ---------------------------------------------------------------------------

// Kernel 1: h[b,d,n] = relu( sum_c W1[d,c]*x[b,c,n] + b1[d] )
// grid (N/16, B), block 256 = 8 waves; wave w -> d-tile [16w, 16w+16)
__global__ __launch_bounds__(256) void k1_gemm_relu(
    const float* __restrict__ x, const float* __restrict__ W1,
    const float* __restrict__ b1, float* __restrict__ h)
{
    const int n0   = blockIdx.x * 16;
    const int b    = blockIdx.y;
    const int lane = threadIdx.x & 31;
    const int d0   = (threadIdx.x >> 5) * 16;
    const int col  = lane & 15;
    const int g    = lane >> 4;
    const float* xb = x + (size_t)b * CC * NN;

    v8f acc = {};
    for (int k = 0; k < CC; k += 4) {
        v2f a, bm;
        a.x  = W1[(d0 + col) * CC + k + 2 * g + 0];
        a.y  = W1[(d0 + col) * CC + k + 2 * g + 1];
        bm.x = xb[(size_t)(k + 2 * g + 0) * NN + n0 + col];
        bm.y = xb[(size_t)(k + 2 * g + 1) * NN + n0 + col];
        acc = __builtin_amdgcn_wmma_f32_16x16x4_f32(
            false, a, false, bm, (short)0, acc, false, false);
    }
#pragma unroll
    for (int v = 0; v < 8; ++v) {
        int d = d0 + v + 8 * g;
        float val = fmaxf(acc[v] + b1[d], 0.0f);
        h[((size_t)b * CC + d) * NN + n0 + col] = val;
    }
}

// Kernel 2: xjmax[b,d,n] = relu( max_j( sum_c Wp[d,c]*h[b,c,idx[b,n,j]] ) + bp[d] )
// (max commutes with +bias and relu since both are monotone)
// grid (N, B), block 256 = 8 waves; the 16 neighbors are exactly one B tile.
__global__ __launch_bounds__(256) void k2_gather_gemm_max(
    const float* __restrict__ h, const int* __restrict__ idx,
    const float* __restrict__ Wp, const float* __restrict__ bp,
    float* __restrict__ xjmax)
{
    const int n    = blockIdx.x;
    const int b    = blockIdx.y;
    const int lane = threadIdx.x & 31;
    const int d0   = (threadIdx.x >> 5) * 16;
    const int col  = lane & 15;
    const int g    = lane >> 4;
    const float* hb = h + (size_t)b * CC * NN;
    const int nbr = idx[((size_t)b * NN + n) * KK + col];   // neighbor for column j=col

    v8f acc = {};
    for (int k = 0; k < CC; k += 4) {
        v2f a, bm;
        a.x  = Wp[(d0 + col) * CC + k + 2 * g + 0];
        a.y  = Wp[(d0 + col) * CC + k + 2 * g + 1];
        bm.x = hb[(size_t)(k + 2 * g + 0) * NN + nbr];      // gathered column
        bm.y = hb[(size_t)(k + 2 * g + 1) * NN + nbr];
        acc = __builtin_amdgcn_wmma_f32_16x16x4_f32(
            false, a, false, bm, (short)0, acc, false, false);
    }
    // Max over the 16 neighbor columns: columns live across 16-lane halves.
#pragma unroll
    for (int v = 0; v < 8; ++v) {
        float val = acc[v];
        val = fmaxf(val, __shfl_xor(val, 1, 16));
        val = fmaxf(val, __shfl_xor(val, 2, 16));
        val = fmaxf(val, __shfl_xor(val, 4, 16));
        val = fmaxf(val, __shfl_xor(val, 8, 16));
        int d = d0 + v + 8 * g;
        val = fmaxf(val + bp[d], 0.0f);
        if (col == 0)
            xjmax[((size_t)b * CC + d) * NN + n] = val;
    }
}

// Kernel 3: out[b,d,n] = relu( sum_{c<256} W2[d,c]*cat[c,n] + b2[d] ),
// cat = [h ; xjmax] (never materialized). grid (N/16, B), block 512 = 16 waves.
__global__ __launch_bounds__(512) void k3_gemm_relu(
    const float* __restrict__ h, const float* __restrict__ xjmax,
    const float* __restrict__ W2, const float* __restrict__ b2,
    float* __restrict__ out)
{
    const int n0   = blockIdx.x * 16;
    const int b    = blockIdx.y;
    const int lane = threadIdx.x & 31;
    const int d0   = (threadIdx.x >> 5) * 16;   // 16 waves -> d0 in [0,256)
    const int col  = lane & 15;
    const int g    = lane >> 4;
    const float* hb = h     + (size_t)b * CC * NN;
    const float* xb = xjmax + (size_t)b * CC * NN;

    v8f acc = {};
    for (int k = 0; k < 2 * CC; k += 4) {
        const float* src = (k < CC) ? hb : xb;   // uniform per k-step
        const int kr = k & (CC - 1);
        v2f a, bm;
        a.x  = W2[(d0 + col) * (2 * CC) + k + 2 * g + 0];
        a.y  = W2[(d0 + col) * (2 * CC) + k + 2 * g + 1];
        bm.x = src[(size_t)(kr + 2 * g + 0) * NN + n0 + col];
        bm.y = src[(size_t)(kr + 2 * g + 1) * NN + n0 + col];
        acc = __builtin_amdgcn_wmma_f32_16x16x4_f32(
            false, a, false, bm, (short)0, acc, false, false);
    }
#pragma unroll
    for (int v = 0; v < 8; ++v) {
        int d = d0 + v + 8 * g;
        float val = fmaxf(acc[v] + b2[d], 0.0f);
        out[((size_t)b * COUT + d) * NN + n0 + col] = val;
    }
}

extern "C" void kernel_launch(void* const* d_in, const int* in_sizes, int n_in,
                              void* d_out, int out_size, void* d_ws, size_t ws_size,
                              hipStream_t stream) {
    const float* x    = (const float*)d_in[0];
    const int*   edge = (const int*)  d_in[1];   // (2,B,N,K): use first half = edge_index[0]
    const float* W1   = (const float*)d_in[2];
    const float* b1   = (const float*)d_in[3];
    const float* Wp   = (const float*)d_in[4];
    const float* bp   = (const float*)d_in[5];
    const float* W2   = (const float*)d_in[6];
    const float* b2   = (const float*)d_in[7];
    float* out = (float*)d_out;

    float* h     = (float*)d_ws;                     // B*C*N floats = 40.96 MB
    float* xjmax = h + (size_t)BB * CC * NN;         // B*C*N floats = 40.96 MB

    dim3 g1(NN / 16, BB);
    k1_gemm_relu<<<g1, 256, 0, stream>>>(x, W1, b1, h);

    dim3 g2(NN, BB);
    k2_gather_gemm_max<<<g2, 256, 0, stream>>>(h, edge, Wp, bp, xjmax);

    dim3 g3(NN / 16, BB);
    k3_gemm_relu<<<g3, 512, 0, stream>>>(h, xjmax, W2, b2, out);
}